// GAT18_57543971832607
// MI455X (gfx1250) — compile-verified
//
#include <hip/hip_runtime.h>
#include <math.h>

#define NNODES 3840

typedef __attribute__((ext_vector_type(2))) float v2f;
typedef __attribute__((ext_vector_type(8))) float v8f;

// ---------------------------------------------------------------------------
// Row-major fp32 WMMA GEMM:  C[M,N] = A[M,K] @ B[K,N]   (no bias)
// One wave computes a 16 x (16*NT) tile with NT accumulators, reusing the
// A fragment NT times per k-step (V_WMMA_F32_16X16X4_F32, exact fp32).
// Wave mapping: w = blockIdx.x*blockDim.y + threadIdx.y
//   colGroup = w & ((1<<cgLog2)-1), rowTile = w >> cgLog2  (exact division,
//   no guards -> EXEC all ones for every WMMA).
// A-fragment (16x4 f32): lanes 0-15 = rows; a[0]=A[m][k0+2*lh], a[1]=+1.
// B-fragment (4x16):     lane = col;     b[0]=B[k0+2*lh][n], b[1]=+1.
// D (16x16):             vgpr i + half -> row m0+i+8*lh, lane&15 -> col.
// ---------------------------------------------------------------------------
template <int NT>
__global__ void wmma_gemm_rm(const float* __restrict__ A, const float* __restrict__ Bm,
                             float* __restrict__ C, int K, int N, int cgLog2) {
  int w = blockIdx.x * blockDim.y + threadIdx.y;
  int cg = w & ((1 << cgLog2) - 1);
  int rowTile = w >> cgLog2;

  int lane = threadIdx.x;
  int lh = lane >> 4;
  int lm = lane & 15;
  int m0 = rowTile * 16;
  int n0 = cg * (16 * NT);

  const float* aRow = A + (size_t)(m0 + lm) * K;   // + k (contiguous pair)
  const float* bB   = Bm + n0 + lm;                // + k*N + 16*t

  v8f acc[NT];
  #pragma unroll
  for (int t = 0; t < NT; ++t) acc[t] = (v8f){};

  for (int k0 = 0; k0 < K; k0 += 4) {
    int ka = k0 + 2 * lh;
    v2f a;
    a[0] = aRow[ka];
    a[1] = aRow[ka + 1];
    #pragma unroll
    for (int t = 0; t < NT; ++t) {
      v2f b;
      b[0] = bB[ka * N + 16 * t];
      b[1] = bB[(ka + 1) * N + 16 * t];
      acc[t] = __builtin_amdgcn_wmma_f32_16x16x4_f32(false, a, false, b,
                                                     (short)0, acc[t], false, false);
    }
  }

  #pragma unroll
  for (int t = 0; t < NT; ++t) {
    int n = n0 + 16 * t + lm;
    #pragma unroll
    for (int i = 0; i < 8; ++i)
      C[(size_t)(m0 + i + 8 * lh) * N + n] = acc[t][i];
  }
}

// ---------------------------------------------------------------------------
// cls head, all strides hardcoded:
//   A[m=hw][k=c] = X[b*122880 + c*3840 + hw]   (raw .view reshape)
//   Out[b][o][hw] = sum_c A[hw][c]*W[c][o] + bias[o]
// M=3840, N=128, K=32; wave tile 16x64 (NT=4); 2 column groups.
// C is contiguous along m -> float4 stores.
// ---------------------------------------------------------------------------
__global__ void wmma_gemm_cls(const float* __restrict__ X, const float* __restrict__ W,
                              const float* __restrict__ bias, float* __restrict__ Out) {
  const int b = blockIdx.z;
  int w = blockIdx.x * blockDim.y + threadIdx.y;   // 0..479
  int cg = w & 1;                                  // 2 col groups of 64
  int rowTile = w >> 1;                            // 240 row tiles

  int lane = threadIdx.x;
  int lh = lane >> 4;
  int lm = lane & 15;
  int m0 = rowTile * 16;
  int n0 = cg * 64;

  const float* Xb = X + (size_t)b * (32 * NNODES);

  v8f acc[4];
  #pragma unroll
  for (int t = 0; t < 4; ++t) acc[t] = (v8f){};

  #pragma unroll
  for (int k0 = 0; k0 < 32; k0 += 4) {
    int ka = k0 + 2 * lh;
    v2f a;
    a[0] = Xb[ka * NNODES + m0 + lm];
    a[1] = Xb[(ka + 1) * NNODES + m0 + lm];
    #pragma unroll
    for (int t = 0; t < 4; ++t) {
      v2f bb;
      bb[0] = W[ka * 128 + n0 + 16 * t + lm];
      bb[1] = W[(ka + 1) * 128 + n0 + 16 * t + lm];
      acc[t] = __builtin_amdgcn_wmma_f32_16x16x4_f32(false, a, false, bb,
                                                     (short)0, acc[t], false, false);
    }
  }

  float* Ob = Out + (size_t)b * (128 * NNODES);
  #pragma unroll
  for (int t = 0; t < 4; ++t) {
    int n = n0 + 16 * t + lm;
    float bv = bias[n];
    float* p = Ob + (size_t)n * NNODES + m0 + 8 * lh;
    float4 lo = make_float4(acc[t][0] + bv, acc[t][1] + bv, acc[t][2] + bv, acc[t][3] + bv);
    float4 hi = make_float4(acc[t][4] + bv, acc[t][5] + bv, acc[t][6] + bv, acc[t][7] + bv);
    *(float4*)(p)     = lo;
    *(float4*)(p + 4) = hi;
  }
}

// ---------------------------------------------------------------------------
// Edge / softmax kernels
// ---------------------------------------------------------------------------
__device__ __forceinline__ void atomicMaxF32(float* addr, float val) {
  unsigned int* ua = reinterpret_cast<unsigned int*>(addr);
  unsigned int old = __float_as_uint(*addr);
  while (__uint_as_float(old) < val) {
    unsigned int assumed = old;
    old = atomicCAS(ua, assumed, __float_as_uint(val));
    if (old == assumed) break;
  }
}

__global__ void k_init_stats(float* __restrict__ smax, float* __restrict__ ssum, long long n) {
  long long i = (long long)blockIdx.x * blockDim.x + threadIdx.x;
  if (i < n) { smax[i] = -INFINITY; ssum[i] = 0.0f; }
}

__global__ void k_zero(float* __restrict__ p, long long n) {
  long long i = (long long)blockIdx.x * blockDim.x + threadIdx.x;
  if (i < n) p[i] = 0.0f;
}

// pass 1: logits[b,e,h] = attn[h,:] . leaky_relu(hs[src]+hd[dst]); atomic max per (dst,h)
__global__ void k_edge_logits(const float* __restrict__ hs, const float* __restrict__ hd,
                              const float* __restrict__ attn,
                              const int* __restrict__ src, const int* __restrict__ dst,
                              float* __restrict__ logits, float* __restrict__ smax,
                              int E, int H, int O, int D, long long total) {
  long long i = (long long)blockIdx.x * blockDim.x + threadIdx.x;
  if (i >= total) return;
  int h = (int)(i % H);
  long long be = i / H;
  int e = (int)(be % E);
  int b = (int)(be / E);
  int s = src[e], t = dst[e];

  const float4* ps = (const float4*)(hs + ((long long)b * NNODES + s) * D + h * O);
  const float4* pd = (const float4*)(hd + ((long long)b * NNODES + t) * D + h * O);
  const float4* pa = (const float4*)(attn + h * O);

  float acc = 0.0f;
  int q4 = O >> 2;
  for (int q = 0; q < q4; ++q) {
    float4 sv = ps[q], dv = pd[q], av = pa[q];
    float v;
    v = sv.x + dv.x; acc += av.x * (v > 0.0f ? v : 0.2f * v);
    v = sv.y + dv.y; acc += av.y * (v > 0.0f ? v : 0.2f * v);
    v = sv.z + dv.z; acc += av.z * (v > 0.0f ? v : 0.2f * v);
    v = sv.w + dv.w; acc += av.w * (v > 0.0f ? v : 0.2f * v);
  }
  logits[i] = acc;
  atomicMaxF32(&smax[((long long)b * NNODES + t) * H + h], acc);
}

// pass 2: logits <- exp(logit - max[dst]); atomic add into denominator
__global__ void k_edge_expsum(const int* __restrict__ dst,
                              float* __restrict__ logits,
                              const float* __restrict__ smax, float* __restrict__ ssum,
                              int E, int H, long long total) {
  long long i = (long long)blockIdx.x * blockDim.x + threadIdx.x;
  if (i >= total) return;
  int h = (int)(i % H);
  long long be = i / H;
  int e = (int)(be % E);
  int b = (int)(be / E);
  int t = dst[e];
  long long si = ((long long)b * NNODES + t) * H + h;
  float ex = expf(logits[i] - smax[si]);
  logits[i] = ex;
  atomicAdd(&ssum[si], ex);
}

// pass 3: acc[dst] += (ex/den) * hs[src], float4 granularity
__global__ void k_edge_agg(const float* __restrict__ hs,
                           const int* __restrict__ src, const int* __restrict__ dst,
                           const float* __restrict__ logits, const float* __restrict__ ssum,
                           float* __restrict__ acc,
                           int E, int H, int O, int D, long long total) {
  long long i = (long long)blockIdx.x * blockDim.x + threadIdx.x;
  if (i >= total) return;
  int q4 = D >> 2;
  int q = (int)(i % q4);
  long long be = i / q4;
  int e = (int)(be % E);
  int b = (int)(be / E);
  int d0 = q * 4;
  int h = d0 / O;          // O % 4 == 0 so a float4 never straddles heads
  int s = src[e], t = dst[e];

  float ex  = logits[((long long)b * E + e) * H + h];
  float den = ssum[((long long)b * NNODES + t) * H + h];
  float alpha = ex / den;

  float4 v = *(const float4*)(hs + ((long long)b * NNODES + s) * D + d0);
  float* po = acc + ((long long)b * NNODES + t) * D + d0;
  atomicAdd(po + 0, alpha * v.x);
  atomicAdd(po + 1, alpha * v.y);
  atomicAdd(po + 2, alpha * v.z);
  atomicAdd(po + 3, alpha * v.w);
}

// finalize: out = elu(acc + res + bias)   (in-place safe, res may be null)
__global__ void k_finalize(float* __restrict__ acc, const float* __restrict__ res,
                           const float* __restrict__ bias, float* __restrict__ outp,
                           int D, long long total) {
  long long i = (long long)blockIdx.x * blockDim.x + threadIdx.x;
  if (i >= total) return;
  int d = (int)(i % D);
  float v = acc[i] + (res ? res[i] : 0.0f) + bias[d];
  outp[i] = v > 0.0f ? v : expm1f(v);
}

// ---------------------------------------------------------------------------
// Host side
// ---------------------------------------------------------------------------
static inline int cdiv_i(long long a, int b) { return (int)((a + b - 1) / b); }

// M multiple of 64, N in {256,128,32}, K multiple of 8
static void gemm_rm(hipStream_t st, const float* A, const float* Bm, float* C,
                    int M, int N, int K) {
  int mTiles = M / 16;
  if (N >= 64) {
    int cgroups = N / 64;                       // NT=4 -> 64 cols per wave
    int cgLog2 = (cgroups == 4) ? 2 : (cgroups == 2) ? 1 : 0;
    int waves = mTiles << cgLog2;
    dim3 blk(32, 4), grd(waves / 4, 1, 1);
    wmma_gemm_rm<4><<<grd, blk, 0, st>>>(A, Bm, C, K, N, cgLog2);
  } else {                                      // N == 32 -> NT=2, one group
    int waves = mTiles;
    dim3 blk(32, 4), grd(waves / 4, 1, 1);
    wmma_gemm_rm<2><<<grd, blk, 0, st>>>(A, Bm, C, K, N, 0);
  }
}

static void run_gat_edges(hipStream_t st,
                          const float* hs, const float* hd,
                          const float* attn, const float* bias, const float* res,
                          float* acc, float* lg, float* smax, float* ssum,
                          const int* src, const int* dst,
                          int B, int E, int H, int O) {
  const int D = H * O;
  const long long nStats = (long long)B * NNODES * H;
  const long long nAcc   = (long long)B * NNODES * D;
  const long long nEH    = (long long)B * E * H;
  const long long nAg    = (long long)B * E * (D / 4);

  k_init_stats<<<cdiv_i(nStats, 256), 256, 0, st>>>(smax, ssum, nStats);
  k_zero<<<cdiv_i(nAcc, 256), 256, 0, st>>>(acc, nAcc);
  k_edge_logits<<<cdiv_i(nEH, 256), 256, 0, st>>>(hs, hd, attn, src, dst, lg, smax,
                                                  E, H, O, D, nEH);
  k_edge_expsum<<<cdiv_i(nEH, 256), 256, 0, st>>>(dst, lg, smax, ssum, E, H, nEH);
  k_edge_agg<<<cdiv_i(nAg, 256), 256, 0, st>>>(hs, src, dst, lg, ssum, acc,
                                               E, H, O, D, nAg);
  k_finalize<<<cdiv_i(nAcc, 256), 256, 0, st>>>(acc, res, bias, acc, D, nAcc);
}

extern "C" void kernel_launch(void* const* d_in, const int* in_sizes, int n_in,
                              void* d_out, int out_size, void* d_ws, size_t ws_size,
                              hipStream_t stream) {
  const float* feat  = (const float*)d_in[0];
  const int*   src   = (const int*)d_in[1];
  const int*   dst   = (const int*)d_in[2];
  const float* Wsrc0 = (const float*)d_in[3];
  const float* Wdst0 = (const float*)d_in[4];
  const float* attn0 = (const float*)d_in[5];
  const float* b0    = (const float*)d_in[6];
  const float* Wsrc1 = (const float*)d_in[7];
  const float* Wdst1 = (const float*)d_in[8];
  const float* attn1 = (const float*)d_in[9];
  const float* b1    = (const float*)d_in[10];
  const float* Wres1 = (const float*)d_in[11];
  const float* Wsrc2 = (const float*)d_in[12];
  const float* Wdst2 = (const float*)d_in[13];
  const float* attn2 = (const float*)d_in[14];
  const float* b2    = (const float*)d_in[15];
  const float* Wres2 = (const float*)d_in[16];
  const float* Wcls  = (const float*)d_in[17];
  const float* bcls  = (const float*)d_in[18];
  float* out = (float*)d_out;

  const int B = in_sizes[0] / (NNODES * 128);
  const int E = in_sizes[1];
  const long long BN = (long long)B * NNODES;

  // workspace layout (floats)
  float* ws    = (float*)d_ws;
  float* slabA = ws;                         // BN*256
  float* slabB = slabA + BN * 256;           // BN*256
  float* slabC = slabB + BN * 256;           // BN*256
  float* lg    = slabC + BN * 256;           // B*E*4
  float* smax  = lg + (long long)B * E * 4;  // B*N*4
  float* ssum  = smax + BN * 4;              // B*N*4

  const int M = (int)BN;

  // ---- layer 0: 128 -> 4x64, no residual ----
  gemm_rm(stream, feat, Wsrc0, slabA, M, 256, 128);
  gemm_rm(stream, feat, Wdst0, slabB, M, 256, 128);
  run_gat_edges(stream, slabA, slabB, attn0, b0, nullptr, slabC, lg, smax, ssum,
                src, dst, B, E, 4, 64);                       // h1 = slabC  [BN,256]

  // ---- layer 1: 256 -> 4x32, residual via Wres1 ----
  float* hs1  = slabA;
  float* res1 = slabA + BN * 128;
  float* hd1  = slabB;
  float* h2   = slabB + BN * 128;
  gemm_rm(stream, slabC, Wsrc1, hs1,  M, 128, 256);
  gemm_rm(stream, slabC, Wdst1, hd1,  M, 128, 256);
  gemm_rm(stream, slabC, Wres1, res1, M, 128, 256);
  run_gat_edges(stream, hs1, hd1, attn1, b1, res1, h2, lg, smax, ssum,
                src, dst, B, E, 4, 32);                       // h2 [BN,128]

  // ---- layer 2: 128 -> 1x32, residual via Wres2 ----
  float* hs2  = slabC;
  float* hd2  = slabC + BN * 32;
  float* res2 = slabC + 2 * BN * 32;
  float* h3   = slabC + 3 * BN * 32;
  gemm_rm(stream, h2, Wsrc2, hs2,  M, 32, 128);
  gemm_rm(stream, h2, Wdst2, hd2,  M, 32, 128);
  gemm_rm(stream, h2, Wres2, res2, M, 32, 128);
  run_gat_edges(stream, hs2, hd2, attn2, b2, res2, h3, lg, smax, ssum,
                src, dst, B, E, 1, 32);                       // h3 [BN,32]

  // ---- cls head (raw-view reshape folded into strides) ----
  {
    dim3 blk(32, 4), grd(120, 1, B);   // 480 waves per batch sample
    wmma_gemm_cls<<<grd, blk, 0, stream>>>(h3, Wcls, bcls, out);
  }
}